// GIN_79035988181207
// MI455X (gfx1250) — compile-verified
//
#include <hip/hip_runtime.h>

#define N_NODES 50000
#define N_EDGES 800000
#define HID 96
#define N_LAYERS 3
#define BN_EPS 1e-5f

typedef float v2f __attribute__((ext_vector_type(2)));
typedef float v8f __attribute__((ext_vector_type(8)));

// ---------------- embedding gather: x[n,:] = emb[x_idx[n],:] ----------------
__global__ void k_gather(const int* __restrict__ x_idx,
                         const float* __restrict__ emb,
                         float* __restrict__ x) {
  int t = blockIdx.x * blockDim.x + threadIdx.x;        // over N_NODES * 24 float4 chunks
  if (t >= N_NODES * (HID / 4)) return;
  int n = t / (HID / 4);
  int c = (t % (HID / 4)) * 4;
  int v = x_idx[n];
  *(float4*)(x + (size_t)n * HID + c) = *(const float4*)(emb + (size_t)v * HID + c);
}

// ---------------- agg = x (copy), then scatter-add over edges ----------------
__global__ void k_copy4(const float* __restrict__ src, float* __restrict__ dst, long n4) {
  long t = (long)blockIdx.x * blockDim.x + threadIdx.x;
  if (t < n4) ((float4*)dst)[t] = ((const float4*)src)[t];
}

__global__ void k_scatter(const int* __restrict__ esrc,
                          const int* __restrict__ edst,
                          const float* __restrict__ x,
                          float* __restrict__ agg) {
  long t = (long)blockIdx.x * blockDim.x + threadIdx.x; // over N_EDGES * 24 chunks
  if (t >= (long)N_EDGES * (HID / 4)) return;
  int e = (int)(t / (HID / 4));
  int c = (int)(t % (HID / 4)) * 4;
  int s = esrc[e], d = edst[e];
  float4 v = *(const float4*)(x + (size_t)s * HID + c);
  float* o = agg + (size_t)d * HID + c;
  atomicAdd(o + 0, v.x);
  atomicAdd(o + 1, v.y);
  atomicAdd(o + 2, v.z);
  atomicAdd(o + 3, v.w);
}

// ---------------- fused MLP + BN, one block = 16 nodes, 6 waves ----------------
// Wave `w` computes the 16x16 output tile covering columns [16w, 16w+16).
// A(16x4 f32) frag: lane m=lane&15 is row M; kb=(lane>>4)*2 selects K pair.
// C/D(16x16 f32): vgpr v -> row = v + 8*(lane>>4), col = 16w + (lane&15).
__global__ __launch_bounds__(192) void k_mlp(
    const float* __restrict__ agg,
    const float* __restrict__ W1, const float* __restrict__ b1,
    const float* __restrict__ W2, const float* __restrict__ b2,
    const float* __restrict__ gamma, const float* __restrict__ beta,
    const float* __restrict__ rmean, const float* __restrict__ rvar,
    float* __restrict__ xout,      // contiguous [N,96] (next layer input)
    float* __restrict__ xs)        // pre-offset slice of [N,288]
{
  __shared__ float sA[16 * HID];
  __shared__ float sH[16 * HID];
  const int tid  = threadIdx.x;
  const int wave = tid >> 5;
  const int lane = tid & 31;
  const long rowBase = (long)blockIdx.x * 16;

  // stage 16x96 A tile into LDS (float4 vectorized)
  for (int i = tid; i < 16 * (HID / 4); i += 192) {
    int r = i / (HID / 4), c = (i % (HID / 4)) * 4;
    *(float4*)(sA + r * HID + c) = *(const float4*)(agg + (rowBase + r) * HID + c);
  }
  __syncthreads();

  const int m   = lane & 15;
  const int kb  = (lane >> 4) * 2;
  const int col = wave * 16 + m;
  const int rh  = (lane >> 4) * 8;

  // ---- GEMM1: h = relu(agg @ W1 + b1) ----
  v8f c1 = {};
#pragma unroll
  for (int k = 0; k < HID; k += 4) {
    v2f a, b;
    a.x = sA[m * HID + k + kb];
    a.y = sA[m * HID + k + kb + 1];
    b.x = W1[(k + kb) * HID + col];
    b.y = W1[(k + kb + 1) * HID + col];
    c1 = __builtin_amdgcn_wmma_f32_16x16x4_f32(false, a, false, b, (short)0, c1, false, false);
  }
  {
    float bias = b1[col];
#pragma unroll
    for (int v = 0; v < 8; ++v) {
      float hv = c1[v] + bias;
      sH[(rh + v) * HID + col] = hv > 0.f ? hv : 0.f;
    }
  }
  __syncthreads();

  // ---- GEMM2: h2 = relu(h @ W2 + b2); then BN ----
  v8f c2 = {};
#pragma unroll
  for (int k = 0; k < HID; k += 4) {
    v2f a, b;
    a.x = sH[m * HID + k + kb];
    a.y = sH[m * HID + k + kb + 1];
    b.x = W2[(k + kb) * HID + col];
    b.y = W2[(k + kb + 1) * HID + col];
    c2 = __builtin_amdgcn_wmma_f32_16x16x4_f32(false, a, false, b, (short)0, c2, false, false);
  }
  {
    float bias = b2[col];
    float g = gamma[col], bt = beta[col], mu = rmean[col];
    float inv = rsqrtf(rvar[col] + BN_EPS);
#pragma unroll
    for (int v = 0; v < 8; ++v) {
      float hv = c2[v] + bias;
      hv = hv > 0.f ? hv : 0.f;
      float xv = g * (hv - mu) * inv + bt;
      long row = rowBase + rh + v;
      xout[row * HID + col] = xv;
      xs[row * (3 * HID) + col] = xv;
    }
  }
}

// ---------------- lin1: hid = relu(xs[ N x 288 ] @ lin1_w + lin1_b) ----------------
__global__ __launch_bounds__(192) void k_lin1(
    const float* __restrict__ xs,      // [N,288]
    const float* __restrict__ w,       // [288,96]
    const float* __restrict__ bias_v,  // [96]
    float* __restrict__ hid)           // [N,96]
{
  __shared__ float sA[16 * 3 * HID];   // 18 KB
  const int tid  = threadIdx.x;
  const int wave = tid >> 5;
  const int lane = tid & 31;
  const long rowBase = (long)blockIdx.x * 16;
  const int K = 3 * HID;

  for (int i = tid; i < 16 * (K / 4); i += 192) {
    int r = i / (K / 4), c = (i % (K / 4)) * 4;
    *(float4*)(sA + r * K + c) = *(const float4*)(xs + (rowBase + r) * K + c);
  }
  __syncthreads();

  const int m   = lane & 15;
  const int kb  = (lane >> 4) * 2;
  const int col = wave * 16 + m;
  const int rh  = (lane >> 4) * 8;

  v8f c = {};
#pragma unroll
  for (int k = 0; k < 3 * HID; k += 4) {
    v2f a, b;
    a.x = sA[m * K + k + kb];
    a.y = sA[m * K + k + kb + 1];
    b.x = w[(k + kb) * HID + col];
    b.y = w[(k + kb + 1) * HID + col];
    c = __builtin_amdgcn_wmma_f32_16x16x4_f32(false, a, false, b, (short)0, c, false, false);
  }
  float bias = bias_v[col];
#pragma unroll
  for (int v = 0; v < 8; ++v) {
    float hv = c[v] + bias;
    hid[(rowBase + rh + v) * HID + col] = hv > 0.f ? hv : 0.f;
  }
}

// ---------------- lin2: out[N,2] = hid @ lin2_w + lin2_b ----------------
__global__ void k_lin2(const float* __restrict__ hid,
                       const float* __restrict__ w,   // [96,2]
                       const float* __restrict__ b,   // [2]
                       float* __restrict__ out) {
  int t = blockIdx.x * blockDim.x + threadIdx.x;       // N_NODES * 2
  if (t >= N_NODES * 2) return;
  int n = t >> 1, j = t & 1;
  float acc = b[j];
  const float* hr = hid + (size_t)n * HID;
#pragma unroll 8
  for (int h = 0; h < HID; ++h) acc += hr[h] * w[h * 2 + j];
  out[t] = acc;
}

extern "C" void kernel_launch(void* const* d_in, const int* in_sizes, int n_in,
                              void* d_out, int out_size, void* d_ws, size_t ws_size,
                              hipStream_t stream) {
  (void)in_sizes; (void)n_in; (void)out_size; (void)ws_size;
  const int*   x_idx = (const int*)d_in[0];
  const int*   edge  = (const int*)d_in[1];
  const float* emb   = (const float*)d_in[2];
  const float* W1    = (const float*)d_in[3];
  const float* b1    = (const float*)d_in[4];
  const float* W2    = (const float*)d_in[5];
  const float* b2    = (const float*)d_in[6];
  const float* gam   = (const float*)d_in[7];
  const float* bet   = (const float*)d_in[8];
  const float* rme   = (const float*)d_in[9];
  const float* rva   = (const float*)d_in[10];
  const float* l1w   = (const float*)d_in[11];
  const float* l1b   = (const float*)d_in[12];
  const float* l2w   = (const float*)d_in[13];
  const float* l2b   = (const float*)d_in[14];
  float* out = (float*)d_out;

  float* x   = (float*)d_ws;                       // N*96
  float* agg = x   + (size_t)N_NODES * HID;        // N*96
  float* xs  = agg + (size_t)N_NODES * HID;        // N*288
  float* hid = xs  + (size_t)N_NODES * HID * 3;    // N*96

  const int* esrc = edge;
  const int* edst = edge + N_EDGES;

  {
    int total = N_NODES * (HID / 4);
    k_gather<<<(total + 255) / 256, 256, 0, stream>>>(x_idx, emb, x);
  }
  for (int l = 0; l < N_LAYERS; ++l) {
    long n4 = (long)N_NODES * HID / 4;
    k_copy4<<<(int)((n4 + 255) / 256), 256, 0, stream>>>(x, agg, n4);
    long te = (long)N_EDGES * (HID / 4);
    k_scatter<<<(int)((te + 255) / 256), 256, 0, stream>>>(esrc, edst, x, agg);
    k_mlp<<<N_NODES / 16, 192, 0, stream>>>(agg,
        W1 + (size_t)l * HID * HID, b1 + l * HID,
        W2 + (size_t)l * HID * HID, b2 + l * HID,
        gam + l * HID, bet + l * HID, rme + l * HID, rva + l * HID,
        x, xs + l * HID);
  }
  k_lin1<<<N_NODES / 16, 192, 0, stream>>>(xs, l1w, l1b, hid);
  k_lin2<<<(N_NODES * 2 + 255) / 256, 256, 0, stream>>>(hid, l2w, l2b, out);
}